// GNN_1614907703887
// MI455X (gfx1250) — compile-verified
//
#include <hip/hip_runtime.h>
#include <math.h>

typedef float v2f __attribute__((ext_vector_type(2)));
typedef float v8f __attribute__((ext_vector_type(8)));

#define N_NODES 50000
#define N_EDGES 800000
#define E2      (N_EDGES + N_NODES)   /* 850000 edges incl. self loops */
#define G_DIM   128
#define HEADS   4
#define C1      32
#define F1      (HEADS * C1)          /* 128 */
#define F2      64
#define ATT_SLOPE 0.2f
#define LRELU_SLOPE 0.01f
#define BN_EPS 1e-5f

// ---------------------------------------------------------------------------
// f32 WMMA GEMM: C[M,N] = A[M,K] @ B[K,N], all row-major f32.
// M % 16 == 0, N % 16 == 0, K % 4 == 0. One wave (32 threads) per 16x16 tile.
// Uses V_WMMA_F32_16X16X4_F32 (exact f32 -> no precision loss vs reference).
// ---------------------------------------------------------------------------
__global__ __launch_bounds__(32)
void gnn_wmma_gemm_f32(const float* __restrict__ A, const float* __restrict__ B,
                       float* __restrict__ C, int M, int N, int K) {
  const int lane = threadIdx.x;          // 0..31
  const int m0   = blockIdx.x * 16;
  const int n0   = blockIdx.y * 16;
  const int half = lane >> 4;            // 0 or 1 -> which K pair
  const int lid  = lane & 15;            // row (A) / col (B,C)

  v8f acc = {};
  const float* Arow = A + (size_t)(m0 + lid) * K + half * 2;       // A[m][k+kp]
  const float* Bcol = B + (size_t)(half * 2) * N + n0 + lid;       // B[k+kp][n]

  for (int k = 0; k < K; k += 4) {
    v2f a, b;
    a.x = Arow[k];
    a.y = Arow[k + 1];
    b.x = Bcol[(size_t)k * N];
    b.y = Bcol[(size_t)(k + 1) * N];
    acc = __builtin_amdgcn_wmma_f32_16x16x4_f32(
        /*neg_a=*/false, a, /*neg_b=*/false, b,
        /*c_mod=*/(short)0, acc, /*reuse_a=*/false, /*reuse_b=*/false);
  }

#pragma unroll
  for (int v = 0; v < 8; ++v) {
    C[(size_t)(m0 + v + half * 8) * N + n0 + lid] = acc[v];
  }
}

// ---------------------------------------------------------------------------
// Per-node attention coefficients: al_s[n,h] = sum_c xw[n,h,c]*a_src[h,c]
// ---------------------------------------------------------------------------
__global__ void gnn_attn_coeff(const float* __restrict__ xw,
                               const float* __restrict__ asrc,
                               const float* __restrict__ adst,
                               float* __restrict__ als, float* __restrict__ ald,
                               int H, int C) {
  int t = blockIdx.x * blockDim.x + threadIdx.x;
  if (t >= N_NODES * H) return;
  int n = t / H, h = t - n * H;
  const float* x = xw + (size_t)n * H * C + (size_t)h * C;
  float ss = 0.f, sd = 0.f;
  for (int c = 0; c < C; ++c) {
    ss += x[c] * asrc[h * C + c];
    sd += x[c] * adst[h * C + c];
  }
  als[t] = ss;
  ald[t] = sd;
}

// ---------------------------------------------------------------------------
// Init accumulators for one attention pass.
// ---------------------------------------------------------------------------
__global__ void gnn_init_attn(float* __restrict__ agg, int aggN,
                              float* __restrict__ sum, int* __restrict__ mx,
                              int segN) {
  int i = blockIdx.x * blockDim.x + threadIdx.x;
  if (i < aggN) agg[i] = 0.f;
  if (i < segN) { sum[i] = 0.f; mx[i] = (int)0x80000000; }
}

__device__ __forceinline__ void gnn_edge_sd(const int* ei, int e, int& s, int& d) {
  if (e < N_EDGES) { s = ei[e]; d = ei[N_EDGES + e]; }
  else             { s = d = e - N_EDGES; }     // self loop
}

__device__ __forceinline__ int gnn_f2ord(float f) {
  int b = __float_as_int(f);
  return b >= 0 ? b : (b ^ 0x7fffffff);
}
__device__ __forceinline__ float gnn_ord2f(int b) {
  return __int_as_float(b >= 0 ? b : (b ^ 0x7fffffff));
}

// ---------------------------------------------------------------------------
// Pass 1: logits + segment max (order-preserving int key atomicMax).
// ---------------------------------------------------------------------------
__global__ void gnn_edge_logit_max(const int* __restrict__ ei,
                                   const float* __restrict__ als,
                                   const float* __restrict__ ald,
                                   float* __restrict__ logits,
                                   int* __restrict__ mx, int H) {
  int t = blockIdx.x * blockDim.x + threadIdx.x;
  if (t >= E2 * H) return;
  int e = t / H, h = t - e * H, s, d;
  gnn_edge_sd(ei, e, s, d);
  float x = als[s * H + h] + ald[d * H + h];
  x = x > 0.f ? x : ATT_SLOPE * x;               // leaky_relu(.., 0.2)
  logits[t] = x;
  atomicMax(&mx[d * H + h], gnn_f2ord(x));
}

// ---------------------------------------------------------------------------
// Pass 2: exp(logit - max) + segment sum.
// ---------------------------------------------------------------------------
__global__ void gnn_edge_expsum(const int* __restrict__ ei,
                                const float* __restrict__ logits,
                                const int* __restrict__ mx,
                                float* __restrict__ ex,
                                float* __restrict__ sum, int H) {
  int t = blockIdx.x * blockDim.x + threadIdx.x;
  if (t >= E2 * H) return;
  int e = t / H, h = t - e * H, s, d;
  gnn_edge_sd(ei, e, s, d);
  float m = gnn_ord2f(mx[d * H + h]);
  float v = expf(logits[t] - m);
  ex[t] = v;
  atomicAdd(&sum[d * H + h], v);
}

// ---------------------------------------------------------------------------
// Pass 3: alpha * xw[src] scatter-added into agg[dst].
// ---------------------------------------------------------------------------
__global__ void gnn_edge_aggregate(const int* __restrict__ ei,
                                   const float* __restrict__ ex,
                                   const float* __restrict__ sum,
                                   const float* __restrict__ xw,
                                   float* __restrict__ agg, int H, int C) {
  int t = blockIdx.x * blockDim.x + threadIdx.x;
  if (t >= E2 * H) return;
  int e = t / H, h = t - e * H, s, d;
  gnn_edge_sd(ei, e, s, d);
  float alpha = ex[t] / (sum[d * H + h] + 1e-16f);
  const float* xs = xw + (size_t)s * H * C + (size_t)h * C;
  float* ag = agg + (size_t)d * H * C + (size_t)h * C;
  for (int c = 0; c < C; ++c) atomicAdd(&ag[c], alpha * xs[c]);
}

// ---------------------------------------------------------------------------
// BN (train mode, biased var): per-channel mean & 1/sqrt(var+eps).
// One block (256 threads) per channel.
// ---------------------------------------------------------------------------
__global__ __launch_bounds__(256)
void gnn_bn_stats(const float* __restrict__ x, float* __restrict__ mean,
                  float* __restrict__ istd, int F) {
  int j = blockIdx.x;
  __shared__ float s1[256], s2[256];
  float a = 0.f, b = 0.f;
  for (int i = threadIdx.x; i < N_NODES; i += 256) {
    float v = x[(size_t)i * F + j];
    a += v; b += v * v;
  }
  s1[threadIdx.x] = a; s2[threadIdx.x] = b;
  __syncthreads();
  for (int off = 128; off > 0; off >>= 1) {
    if (threadIdx.x < off) {
      s1[threadIdx.x] += s1[threadIdx.x + off];
      s2[threadIdx.x] += s2[threadIdx.x + off];
    }
    __syncthreads();
  }
  if (threadIdx.x == 0) {
    float mu  = s1[0] / (float)N_NODES;
    float var = s2[0] / (float)N_NODES - mu * mu;
    mean[j] = mu;
    istd[j] = rsqrtf(var + BN_EPS);
  }
}

// y = leaky_relu((x-mu)*istd*gamma + beta, 0.01). (Bias before train-BN cancels.)
__global__ void gnn_bn_norm_lrelu(const float* __restrict__ x,
                                  const float* __restrict__ mean,
                                  const float* __restrict__ istd,
                                  const float* __restrict__ g,
                                  const float* __restrict__ bt,
                                  float* __restrict__ y, int F, int total) {
  int i = blockIdx.x * blockDim.x + threadIdx.x;
  if (i >= total) return;
  int j = i % F;
  float v = (x[i] - mean[j]) * istd[j] * g[j] + bt[j];
  y[i] = v > 0.f ? v : LRELU_SLOPE * v;
}

// ---------------------------------------------------------------------------
// Launch sequence
// ---------------------------------------------------------------------------
extern "C" void kernel_launch(void* const* d_in, const int* in_sizes, int n_in,
                              void* d_out, int out_size, void* d_ws, size_t ws_size,
                              hipStream_t stream) {
  const float* X      = (const float*)d_in[0];
  const int*   ei     = (const int*)  d_in[2];
  const float* W1     = (const float*)d_in[3];
  const float* a_src1 = (const float*)d_in[4];
  const float* a_dst1 = (const float*)d_in[5];
  const float* g1     = (const float*)d_in[7];
  const float* beta1  = (const float*)d_in[8];
  const float* W2     = (const float*)d_in[9];
  const float* a_src2 = (const float*)d_in[10];
  const float* a_dst2 = (const float*)d_in[11];
  const float* g2     = (const float*)d_in[13];
  const float* beta2  = (const float*)d_in[14];
  float* out = (float*)d_out;

  // Workspace layout (floats). agg buffer is reused in-place: agg1 -> x1 -> agg2.
  float* ws = (float*)d_ws;
  size_t o = 0;
  float* xw1    = ws + o; o += (size_t)N_NODES * F1;        // 6.4M
  float* agg    = ws + o; o += (size_t)N_NODES * F1;        // 6.4M (agg1/x1/agg2)
  float* als    = ws + o; o += (size_t)N_NODES * HEADS;     // 200k (als1 / als2)
  float* ald    = ws + o; o += (size_t)N_NODES * HEADS;     // 200k
  float* logits = ws + o; o += (size_t)E2 * HEADS;          // 3.4M (l1; l2+ex2)
  float* exbuf  = ws + o; o += (size_t)E2 * HEADS;          // 3.4M (ex1 / xw2)
  float* dsum   = ws + o; o += (size_t)N_NODES * HEADS;     // 200k
  float* mean   = ws + o; o += 128;
  float* istd   = ws + o; o += 128;
  int*   dmax   = (int*)(ws + o); o += (size_t)N_NODES * HEADS;

  const int EH1 = E2 * HEADS;       // 3.4M
  const int A1  = N_NODES * F1;     // 6.4M
  const int S1  = N_NODES * HEADS;  // 200k
  const int A2  = N_NODES * F2;     // 3.2M

  // ---- Layer 1 ----
  gnn_wmma_gemm_f32<<<dim3(N_NODES / 16, F1 / 16), 32, 0, stream>>>(
      X, W1, xw1, N_NODES, F1, G_DIM);
  gnn_attn_coeff<<<(S1 + 255) / 256, 256, 0, stream>>>(
      xw1, a_src1, a_dst1, als, ald, HEADS, C1);
  gnn_init_attn<<<(A1 + 255) / 256, 256, 0, stream>>>(agg, A1, dsum, dmax, S1);
  gnn_edge_logit_max<<<(EH1 + 255) / 256, 256, 0, stream>>>(
      ei, als, ald, logits, dmax, HEADS);
  gnn_edge_expsum<<<(EH1 + 255) / 256, 256, 0, stream>>>(
      ei, logits, dmax, exbuf, dsum, HEADS);
  gnn_edge_aggregate<<<(EH1 + 255) / 256, 256, 0, stream>>>(
      ei, exbuf, dsum, xw1, agg, HEADS, C1);
  gnn_bn_stats<<<F1, 256, 0, stream>>>(agg, mean, istd, F1);
  gnn_bn_norm_lrelu<<<(A1 + 255) / 256, 256, 0, stream>>>(
      agg, mean, istd, g1, beta1, agg, F1, A1);   // in-place: agg -> x1

  // ---- Layer 2 (H=1, C=64; mean over 1 head == identity) ----
  float* x1      = agg;
  float* xw2     = exbuf;                 // reuse ex1 region (3.2M <= 3.4M)
  float* logits2 = logits;                // [0, 850k)
  float* ex2     = logits + E2;           // [850k, 1.7M)
  gnn_wmma_gemm_f32<<<dim3(N_NODES / 16, F2 / 16), 32, 0, stream>>>(
      x1, W2, xw2, N_NODES, F2, F1);
  gnn_attn_coeff<<<(N_NODES + 255) / 256, 256, 0, stream>>>(
      xw2, a_src2, a_dst2, als, ald, 1, F2);
  gnn_init_attn<<<(A2 + 255) / 256, 256, 0, stream>>>(agg, A2, dsum, dmax, N_NODES);
  gnn_edge_logit_max<<<(E2 + 255) / 256, 256, 0, stream>>>(
      ei, als, ald, logits2, dmax, 1);
  gnn_edge_expsum<<<(E2 + 255) / 256, 256, 0, stream>>>(
      ei, logits2, dmax, ex2, dsum, 1);
  gnn_edge_aggregate<<<(E2 + 255) / 256, 256, 0, stream>>>(
      ei, ex2, dsum, xw2, agg, 1, F2);
  gnn_bn_stats<<<F2, 256, 0, stream>>>(agg, mean, istd, F2);
  gnn_bn_norm_lrelu<<<(A2 + 255) / 256, 256, 0, stream>>>(
      agg, mean, istd, g2, beta2, out, F2, A2);
  (void)in_sizes; (void)n_in; (void)out_size; (void)ws_size;
}